// PointRendSemSegHead_52226802320026
// MI455X (gfx1250) — compile-verified
//
#include <hip/hip_runtime.h>
#include <math.h>

typedef __attribute__((ext_vector_type(2))) float v2f;
typedef __attribute__((ext_vector_type(8))) float v8f;

#define NIMG 4
#define NIN 128
#define NCLS 3
#define KDIM 131   // NIN + NCLS
#define KPAD 132   // padded K (row 131 = zeros)
#define PSEL 8192
#define PTILE 16

// ---------------------------------------------------------------------------
// monotone float -> uint key (ascending): larger float -> larger key
__device__ __forceinline__ unsigned fkey(float f) {
    unsigned u = __float_as_uint(f);
    return (u & 0x80000000u) ? ~u : (u | 0x80000000u);
}

// bilinear sample with zero padding (grid_sample semantics, half-pixel centers)
__device__ __forceinline__ float sample_bilin_zero(const float* __restrict__ plane,
                                                   int Hh, int Ww, float px, float py) {
    float x = px * (float)Ww - 0.5f;
    float y = py * (float)Hh - 0.5f;
    int x0 = (int)floorf(x), y0 = (int)floorf(y);
    float wx = x - (float)x0, wy = y - (float)y0;
    float acc = 0.f;
#pragma unroll
    for (int dy = 0; dy < 2; ++dy) {
        int yi = y0 + dy;
        if (yi < 0 || yi >= Hh) continue;
        float wyv = dy ? wy : (1.f - wy);
#pragma unroll
        for (int dx = 0; dx < 2; ++dx) {
            int xi = x0 + dx;
            if (xi < 0 || xi >= Ww) continue;
            float wxv = dx ? wx : (1.f - wx);
            acc += wyv * wxv * plane[(size_t)yi * Ww + xi];
        }
    }
    return acc;
}

// ---------------------------------------------------------------------------
// Kernel 1: fused 2x bilinear upsample (edge clamp, half-pixel) + uncertainty
__global__ void upsample_unc_kernel(const float* __restrict__ semIn,
                                    float* __restrict__ semOut,
                                    float* __restrict__ unc,
                                    int Hin, int Win) {
    const int H = 2 * Hin, W = 2 * Win;
    const int HW = H * W;
    int gid = blockIdx.x * blockDim.x + threadIdx.x;
    if (gid >= NIMG * HW) return;
    int n = gid / HW;
    int p = gid - n * HW;
    int oy = p / W, ox = p - oy * W;

    float fy = (float)oy * 0.5f - 0.25f;
    float fx = (float)ox * 0.5f - 0.25f;
    int y0 = (int)floorf(fy);
    int x0 = (int)floorf(fx);
    float wy = fy - (float)y0;
    float wx = fx - (float)x0;
    int y0c = min(Hin - 1, max(0, y0));
    int y1c = min(Hin - 1, max(0, y0 + 1));
    int x0c = min(Win - 1, max(0, x0));
    int x1c = min(Win - 1, max(0, x0 + 1));

    const float* base = semIn + (size_t)n * NCLS * Hin * Win;
    float v[NCLS];
#pragma unroll
    for (int c = 0; c < NCLS; ++c) {
        const float* f = base + (size_t)c * Hin * Win;
        float v00 = f[(size_t)y0c * Win + x0c];
        float v01 = f[(size_t)y0c * Win + x1c];
        float v10 = f[(size_t)y1c * Win + x0c];
        float v11 = f[(size_t)y1c * Win + x1c];
        float val = (1.f - wy) * ((1.f - wx) * v00 + wx * v01)
                  +        wy  * ((1.f - wx) * v10 + wx * v11);
        v[c] = val;
        semOut[((size_t)n * NCLS + c) * HW + p] = val;
    }
    // top2 - top1 for 3 values
    float m1 = fmaxf(v[0], fmaxf(v[1], v[2]));
    float m3 = fminf(v[0], fminf(v[1], v[2]));
    float m2 = v[0] + v[1] + v[2] - m1 - m3;
    unc[(size_t)n * HW + p] = m2 - m1;
}

// ---------------------------------------------------------------------------
// Kernel 2: per-image exact radix select of the P-th largest uncertainty key.
// One workgroup per image; 4 rounds of 8-bit LDS histograms.
// selOut[img*4] = {threshKey, tiesToTake, cntAbove=0, cntTie=0}
__global__ void select_kernel(const float* __restrict__ unc, int HW, int P,
                              unsigned* __restrict__ selOut) {
    const int img = blockIdx.x;
    const float* u = unc + (size_t)img * HW;
    __shared__ unsigned hist[256];
    __shared__ unsigned s_prefix, s_k;

    unsigned prefix = 0, prefixMask = 0;
    unsigned k = (unsigned)P;

    for (int shift = 24; shift >= 0; shift -= 8) {
        for (int i = threadIdx.x; i < 256; i += blockDim.x) hist[i] = 0;
        __syncthreads();
        for (int i = threadIdx.x; i < HW; i += blockDim.x) {
            unsigned key = fkey(u[i]);
            if ((key & prefixMask) == prefix)
                atomicAdd(&hist[(key >> shift) & 0xFFu], 1u);
        }
        __syncthreads();
        if (threadIdx.x == 0) {
            unsigned cum = 0;
            unsigned pfx = prefix, kk = 0;
            for (int b = 255; b >= 0; --b) {
                unsigned c = hist[b];
                if (cum + c >= k) {
                    pfx = prefix | ((unsigned)b << shift);
                    kk = k - cum;
                    break;
                }
                cum += c;
            }
            s_prefix = pfx;
            s_k = kk;
        }
        __syncthreads();
        prefix = s_prefix;
        k = s_k;
        prefixMask |= (0xFFu << shift);
        __syncthreads();
    }
    if (threadIdx.x == 0) {
        selOut[img * 4 + 0] = prefix;  // exact threshold key
        selOut[img * 4 + 1] = k;       // #ties at threshold to accept
        selOut[img * 4 + 2] = 0;       // cntAbove
        selOut[img * 4 + 3] = 0;       // cntTie
    }
}

// ---------------------------------------------------------------------------
// Kernel 3: compact exactly P indices per image: strictly-above -> [0, P-kTie),
// ties fill [P-kTie, P).
__global__ void compact_kernel(const float* __restrict__ unc, int HW, int P,
                               unsigned* __restrict__ sel, int* __restrict__ idxOut) {
    const int img = blockIdx.y;
    int i = blockIdx.x * blockDim.x + threadIdx.x;
    if (i >= HW) return;
    unsigned T = sel[img * 4 + 0];
    unsigned kTie = sel[img * 4 + 1];
    unsigned key = fkey(unc[(size_t)img * HW + i]);
    if (key > T) {
        unsigned pos = atomicAdd(&sel[img * 4 + 2], 1u);
        idxOut[(size_t)img * P + pos] = i;
    } else if (key == T) {
        unsigned t = atomicAdd(&sel[img * 4 + 3], 1u);
        if (t < kTie) idxOut[(size_t)img * P + (P - kTie + t)] = i;
    }
}

// ---------------------------------------------------------------------------
// WMMA GEMM layer: one wave computes a 16(out) x 16(pts) tile of
// out = relu(W(128x131) * src(131x16) + b).  fp32 WMMA 16x16x4, K padded to 132.
__device__ __forceinline__ void gemm_layer(const float (*src)[PTILE],
                                           float (*dst)[PTILE],
                                           const float* __restrict__ Wm,
                                           const float* __restrict__ Bv,
                                           int wave, int lane) {
    const int mrow = wave * 16 + (lane & 15);        // A-frag row (lanes 0-15 / 16-31 same M)
    const int kOff = (lane & 16) ? 2 : 0;            // lanes 16-31 hold K+2, K+3
    const int ncol = lane & 15;                      // B/C-frag column
    v8f acc = {0.f, 0.f, 0.f, 0.f, 0.f, 0.f, 0.f, 0.f};
    for (int k0 = 0; k0 < KPAD; k0 += 4) {
        int ka = k0 + kOff;
        v2f a, b;
        a.x = (ka     < KDIM) ? Wm[(size_t)mrow * KDIM + ka]     : 0.f;
        a.y = (ka + 1 < KDIM) ? Wm[(size_t)mrow * KDIM + ka + 1] : 0.f;
        b.x = src[ka][ncol];       // row 131 is the zero row
        b.y = src[ka + 1][ncol];
        acc = __builtin_amdgcn_wmma_f32_16x16x4_f32(false, a, false, b,
                                                    (short)0, acc, false, false);
    }
    const int hi = (lane & 16) ? 8 : 0;              // C/D: VGPR r -> M = r (+8 for hi lanes)
#pragma unroll
    for (int r = 0; r < 8; ++r) {
        int m = wave * 16 + r + hi;
        float vv = acc[r] + Bv[m];
        dst[m][ncol] = vv > 0.f ? vv : 0.f;
    }
}

// ---------------------------------------------------------------------------
// Kernel 4: fused point-sample + 3-layer MLP (WMMA) + projection + scatter.
// One block = 16 points, 8 waves (256 threads).
__global__ void __launch_bounds__(256)
point_head_kernel(const float* __restrict__ coarse,    // (N,3,128,128)
                  const float* __restrict__ features,  // (N,128,256,256)
                  const float* __restrict__ w1, const float* __restrict__ b1,
                  const float* __restrict__ w2, const float* __restrict__ b2,
                  const float* __restrict__ w3, const float* __restrict__ b3,
                  const float* __restrict__ wp, const float* __restrict__ bp,
                  const int* __restrict__ idxb,        // (N,P)
                  float* __restrict__ semOut,          // (N,3,H*W)
                  int H, int W, int P) {
    __shared__ float xb[KPAD][PTILE];
    __shared__ float yb[KPAD][PTILE];
    __shared__ float pxs[PTILE], pys[PTILE];
    __shared__ int pidx[PTILE];

    const int tid  = threadIdx.x;
    const int wave = tid >> 5;
    const int lane = tid & 31;
    const int tilesPerImg = P / PTILE;
    const int img    = blockIdx.x / tilesPerImg;
    const int ptBase = (blockIdx.x % tilesPerImg) * PTILE;

    if (tid < PTILE) {
        int id = idxb[(size_t)img * P + ptBase + tid];
        pidx[tid] = id;
        int ix = id % W;
        int iy = id / W;
        pxs[tid] = ((float)ix + 0.5f) / (float)W;
        pys[tid] = ((float)iy + 0.5f) / (float)H;
        xb[KDIM][tid] = 0.f;     // zero K-pad row
        yb[KDIM][tid] = 0.f;
    }
    __syncthreads();

    // fine features: 128 channels x 16 points
    for (int t = tid; t < NIN * PTILE; t += 256) {
        int c = t >> 4, pt = t & 15;
        xb[c][pt] = sample_bilin_zero(features + ((size_t)img * NIN + c) * (256 * 256),
                                      256, 256, pxs[pt], pys[pt]);
    }
    // coarse logits: 3 channels x 16 points (persist in BOTH buffers = re-concat each layer)
    if (tid < NCLS * PTILE) {
        int c = tid >> 4, pt = tid & 15;
        float v = sample_bilin_zero(coarse + ((size_t)img * NCLS + c) * (128 * 128),
                                    128, 128, pxs[pt], pys[pt]);
        xb[NIN + c][pt] = v;
        yb[NIN + c][pt] = v;
    }
    __syncthreads();

    gemm_layer(xb, yb, w1, b1, wave, lane); __syncthreads();
    gemm_layer(yb, xb, w2, b2, wave, lane); __syncthreads();
    gemm_layer(xb, yb, w3, b3, wave, lane); __syncthreads();

    // final projection wp (3x131) by wave 0 only (wave-uniform branch, EXEC all-ones)
    if (wave == 0) {
        const int mrow = lane & 15;
        const int kOff = (lane & 16) ? 2 : 0;
        const int ncol = lane & 15;
        v8f acc = {0.f, 0.f, 0.f, 0.f, 0.f, 0.f, 0.f, 0.f};
        for (int k0 = 0; k0 < KPAD; k0 += 4) {
            int ka = k0 + kOff;
            v2f a, b;
            a.x = (mrow < NCLS && ka     < KDIM) ? wp[(size_t)mrow * KDIM + ka]     : 0.f;
            a.y = (mrow < NCLS && ka + 1 < KDIM) ? wp[(size_t)mrow * KDIM + ka + 1] : 0.f;
            b.x = yb[ka][ncol];
            b.y = yb[ka + 1][ncol];
            acc = __builtin_amdgcn_wmma_f32_16x16x4_f32(false, a, false, b,
                                                        (short)0, acc, false, false);
        }
        if (lane < PTILE) {   // rows 0..2 of D live in acc[0..2] of lanes 0-15
            size_t hw = (size_t)H * W;
            size_t base = (size_t)img * NCLS * hw + (size_t)pidx[lane];
            semOut[base]          = acc[0] + bp[0];
            semOut[base + hw]     = acc[1] + bp[1];
            semOut[base + 2 * hw] = acc[2] + bp[2];
        }
    }
}

// ---------------------------------------------------------------------------
extern "C" void kernel_launch(void* const* d_in, const int* in_sizes, int n_in,
                              void* d_out, int out_size, void* d_ws, size_t ws_size,
                              hipStream_t stream) {
    (void)in_sizes; (void)n_in; (void)out_size; (void)ws_size;
    const float* coarse   = (const float*)d_in[0];
    const float* features = (const float*)d_in[1];
    const float* w1 = (const float*)d_in[2];
    const float* b1 = (const float*)d_in[3];
    const float* w2 = (const float*)d_in[4];
    const float* b2 = (const float*)d_in[5];
    const float* w3 = (const float*)d_in[6];
    const float* b3 = (const float*)d_in[7];
    const float* wp = (const float*)d_in[8];
    const float* bp = (const float*)d_in[9];
    float* out = (float*)d_out;

    // workspace carve-out
    char* ws = (char*)d_ws;
    size_t off = 0;
    float* sem1 = (float*)(ws + off); off += (size_t)NIMG * NCLS * 256 * 256 * sizeof(float);
    float* unc  = (float*)(ws + off); off += (size_t)NIMG * 512 * 512 * sizeof(float);
    int*   idxb = (int*)  (ws + off); off += (size_t)NIMG * PSEL * sizeof(int);
    unsigned* sel = (unsigned*)(ws + off); off += (size_t)NIMG * 4 * sizeof(unsigned);

    const int P = PSEL;

    // ---- step 1: 128x128 -> 256x256, refine into sem1 (workspace) ----
    {
        const int Hin = 128, Win = 128, H = 256, W = 256, HW = H * W;
        upsample_unc_kernel<<<(NIMG * HW + 255) / 256, 256, 0, stream>>>(coarse, sem1, unc, Hin, Win);
        select_kernel<<<NIMG, 256, 0, stream>>>(unc, HW, P, sel);
        dim3 cg((HW + 255) / 256, NIMG);
        compact_kernel<<<cg, 256, 0, stream>>>(unc, HW, P, sel, idxb);
        point_head_kernel<<<NIMG * (P / PTILE), 256, 0, stream>>>(
            coarse, features, w1, b1, w2, b2, w3, b3, wp, bp, idxb, sem1, H, W, P);
    }
    // ---- step 2: 256x256 -> 512x512, refine directly into d_out ----
    {
        const int Hin = 256, Win = 256, H = 512, W = 512, HW = H * W;
        upsample_unc_kernel<<<(NIMG * HW + 255) / 256, 256, 0, stream>>>(sem1, out, unc, Hin, Win);
        select_kernel<<<NIMG, 256, 0, stream>>>(unc, HW, P, sel);
        dim3 cg((HW + 255) / 256, NIMG);
        compact_kernel<<<cg, 256, 0, stream>>>(unc, HW, P, sel, idxb);
        point_head_kernel<<<NIMG * (P / PTILE), 256, 0, stream>>>(
            coarse, features, w1, b1, w2, b2, w3, b3, wp, bp, idxb, out, H, W, P);
    }
}